// AlphaGraphPhysFinal_34600256537261
// MI455X (gfx1250) — compile-verified
//
#include <hip/hip_runtime.h>
#include <hip/hip_bf16.h>

// ---------------- problem constants (match reference) ----------------
static constexpr int NND  = 10000;   // nodes
static constexpr int NE   = 80000;   // edges
static constexpr int INC  = 128;
static constexpr int HID  = 256;
static constexpr int HEADS = 8;
static constexpr int HC   = HEADS * HID;   // 2048
static constexpr float NEG_SLOPE = 0.2f;

typedef __attribute__((ext_vector_type(2))) float v2f;
typedef __attribute__((ext_vector_type(8))) float v8f;

// monotonic float<->uint mapping for atomicMax-based segment max
__device__ __forceinline__ unsigned flip_f(float f) {
  unsigned u = __float_as_uint(f);
  return (u & 0x80000000u) ? ~u : (u | 0x80000000u);
}
__device__ __forceinline__ float unflip_f(unsigned v) {
  unsigned u = (v & 0x80000000u) ? (v ^ 0x80000000u) : ~v;
  return __uint_as_float(u);
}

// ---------------- init: out1 = b_gat broadcast; emax=-inf; denom=0; deg=0 ---
__global__ void k_init1(float* __restrict__ out1, const float* __restrict__ b_gat,
                        unsigned* __restrict__ emax, float* __restrict__ denom,
                        float* __restrict__ deg) {
  long long i = (long long)blockIdx.x * blockDim.x + threadIdx.x;
  if (i < (long long)NND * HC) out1[i] = b_gat[i & (HC - 1)];
  if (i < (long long)NND * HEADS) { emax[i] = 0x007FFFFFu; denom[i] = 0.0f; }
  if (i < NND) deg[i] = 0.0f;
}

// ---------------- generic fp32 WMMA GEMM: C = relu?(A@B (+ A2@B2) + bias) ----
// A:[M,KK] row-major, B:[KK,NNc] row-major, C:[M,NNc]. One wave -> 16x64 strip.
// KK, NNc compile-time so the unrolled inner loop uses immediate load offsets.
// Requires: M%16==0, KK%64==0, NNc%64==0.
template <int KK, int NNc>
__global__ void k_wmma_gemm(const float* __restrict__ A, const float* __restrict__ B,
                            const float* __restrict__ A2, const float* __restrict__ B2,
                            const float* __restrict__ bias, float* __restrict__ C,
                            int doRelu) {
  constexpr int KC = 64;
  __shared__ float lA[16 * (KC + 1)];   // pad to avoid 256B bank-conflict period
  const int lane   = threadIdx.x & 31;
  const int wave   = threadIdx.x >> 5;
  const int nWaves = blockDim.x >> 5;
  const int m0     = blockIdx.x * 16;
  const int col    = blockIdx.y * nWaves + wave;
  const int half   = lane >> 4;
  const int lm     = lane & 15;
  const bool active = (col * 64 < NNc);
  const int n0     = col * 64;

  v8f acc[4] = {};

  for (int phase = 0; phase < 2; ++phase) {
    const float* Ap = phase ? A2 : A;
    const float* Bp = phase ? B2 : B;
    if (!Ap) break;  // uniform across block
    for (int kc = 0; kc < KK; kc += KC) {
      __syncthreads();
      for (int t = threadIdx.x; t < 16 * KC; t += blockDim.x) {
        int r = t >> 6, kk = t & (KC - 1);
        lA[r * (KC + 1) + kk] = Ap[(size_t)(m0 + r) * KK + kc + kk];
      }
      __syncthreads();
      if (active) {
        // one base pointer per chunk; all further offsets are immediates
        const float* bp = Bp + (size_t)(kc + 2 * half) * NNc + n0 + lm;
#pragma unroll
        for (int k = 0; k < KC; k += 4) {
          // A fragment: lane holds A[m=lm][k+2*half], A[m=lm][k+2*half+1]
          v2f a;
          a.x = lA[lm * (KC + 1) + k + 2 * half];
          a.y = lA[lm * (KC + 1) + k + 2 * half + 1];
          const float* bk = bp + k * NNc;   // compile-time multiple of NNc
#pragma unroll
          for (int t = 0; t < 4; ++t) {
            // B fragment: lane holds B[k+2*half][n], B[k+2*half+1][n]
            v2f b;
            b.x = bk[t * 16];
            b.y = bk[NNc + t * 16];
            acc[t] = __builtin_amdgcn_wmma_f32_16x16x4_f32(
                false, a, false, b, (short)0, acc[t], false, false);
          }
        }
      }
    }
  }

  if (active) {
#pragma unroll
    for (int t = 0; t < 4; ++t)
#pragma unroll
      for (int i = 0; i < 8; ++i) {
        int m = m0 + i + 8 * half;     // C/D layout: VGPR i -> row i (+8 for hi half)
        int n = n0 + t * 16 + lm;
        float v = acc[t][i];
        if (bias) v += bias[n];
        if (doRelu) v = fmaxf(v, 0.0f);
        C[(size_t)m * NNc + n] = v;
      }
  }
}

// ---------------- attention coefficients: a_src/a_dst [N,H] -----------------
// one wave per node; 4 lanes per head reduce 256 channels
__global__ void k_attn(const float* __restrict__ h, const float* __restrict__ att_src,
                       const float* __restrict__ att_dst, float* __restrict__ a_src,
                       float* __restrict__ a_dst) {
  int node = blockIdx.x * (blockDim.x >> 5) + (threadIdx.x >> 5);
  if (node >= NND) return;
  int lane = threadIdx.x & 31;
  int head = lane >> 2;
  int cs   = lane & 3;
  const float* hp  = h + (size_t)node * HC + head * HID;
  const float* asp = att_src + head * HID;
  const float* adp = att_dst + head * HID;
  float ss = 0.f, sd = 0.f;
  for (int c = cs; c < HID; c += 4) {
    float v = hp[c];
    ss += v * asp[c];
    sd += v * adp[c];
  }
  ss += __shfl_xor(ss, 1); ss += __shfl_xor(ss, 2);
  sd += __shfl_xor(sd, 1); sd += __shfl_xor(sd, 2);
  if (cs == 0) {
    a_src[node * HEADS + head] = ss;
    a_dst[node * HEADS + head] = sd;
  }
}

// ---------------- edge logits + segment max + degree ------------------------
__global__ void k_logits(const long long* __restrict__ ei, const float* __restrict__ a_src,
                         const float* __restrict__ a_dst, float* __restrict__ elog,
                         unsigned* __restrict__ emax, float* __restrict__ deg) {
  int idx = blockIdx.x * blockDim.x + threadIdx.x;   // over E*HEADS
  if (idx >= NE * HEADS) return;
  int e = idx >> 3, head = idx & 7;
  int s = (int)ei[e];
  int d = (int)ei[NE + e];
  float v = a_src[s * HEADS + head] + a_dst[d * HEADS + head];
  v = (v > 0.f) ? v : NEG_SLOPE * v;
  elog[idx] = v;
  atomicMax(&emax[d * HEADS + head], flip_f(v));
  if (head == 0) atomicAdd(&deg[d], 1.0f);
}

// ---------------- exp(e - max) + segment-sum denominator --------------------
__global__ void k_exp(const long long* __restrict__ ei, float* __restrict__ elog,
                      const unsigned* __restrict__ emax, float* __restrict__ denom) {
  int idx = blockIdx.x * blockDim.x + threadIdx.x;
  if (idx >= NE * HEADS) return;
  int e = idx >> 3, head = idx & 7;
  int d = (int)ei[NE + e];
  float m  = unflip_f(emax[d * HEADS + head]);
  float ex = expf(elog[idx] - m);
  elog[idx] = ex;
  atomicAdd(&denom[d * HEADS + head], ex);
}

// ---------------- weighted message scatter: out1[dst] += alpha * h[src] -----
// one wave per (edge, head); 256 channels looped by lane
__global__ void k_message(const long long* __restrict__ ei, const float* __restrict__ h,
                          const float* __restrict__ eexp, const float* __restrict__ denom,
                          float* __restrict__ out1) {
  int pair = blockIdx.x * (blockDim.x >> 5) + (threadIdx.x >> 5);
  if (pair >= NE * HEADS) return;
  int e = pair >> 3, head = pair & 7;
  int lane = threadIdx.x & 31;
  int s = (int)ei[e];
  int d = (int)ei[NE + e];
  float alpha = eexp[pair] / (denom[d * HEADS + head] + 1e-16f);
  const float* hp = h + (size_t)s * HC + head * HID;
  float*       op = out1 + (size_t)d * HC + head * HID;
  for (int c = lane; c < HID; c += 32) atomicAdd(&op[c], alpha * hp[c]);
}

// ---------------- x1 = relu(out1) in place; zero agg (reuses h buffer) ------
__global__ void k_relu_zero(float* __restrict__ x1, float* __restrict__ agg) {
  long long i = (long long)blockIdx.x * blockDim.x + threadIdx.x;
  if (i < (long long)NND * HC) {
    float v = x1[i];
    x1[i]  = (v > 0.f) ? v : 0.f;
    agg[i] = 0.f;
  }
}

// ---------------- SAGE neighbor sum: agg[dst] += x1[src] --------------------
__global__ void k_sage_scatter(const long long* __restrict__ ei, const float* __restrict__ x1,
                               float* __restrict__ agg) {
  int e = blockIdx.x * (blockDim.x >> 5) + (threadIdx.x >> 5);
  if (e >= NE) return;
  int lane = threadIdx.x & 31;
  int s = (int)ei[e];
  int d = (int)ei[NE + e];
  const float* xp = x1  + (size_t)s * HC;
  float*       ap = agg + (size_t)d * HC;
  for (int c = lane; c < HC; c += 32) atomicAdd(&ap[c], xp[c]);
}

// ---------------- mean: agg /= max(deg,1) -----------------------------------
__global__ void k_sagediv(float* __restrict__ agg, const float* __restrict__ deg) {
  long long i = (long long)blockIdx.x * blockDim.x + threadIdx.x;
  if (i < (long long)NND * HC) {
    float dg = fmaxf(deg[i >> 11], 1.0f);   // HC == 2048 == 1<<11
    agg[i] /= dg;
  }
}

// ---------------- final projection: out = hmid @ W_lin2 + b -----------------
__global__ void k_mlp2(const float* __restrict__ hmid, const float* __restrict__ W,
                       const float* __restrict__ b, float* __restrict__ out) {
  int n = blockIdx.x * blockDim.x + threadIdx.x;
  if (n >= NND) return;
  const float* hp = hmid + (size_t)n * (HID / 2);
  float s = 0.f;
#pragma unroll 4
  for (int c = 0; c < HID / 2; ++c) s += hp[c] * W[c];
  out[n] = s + b[0];
}

// ============================================================================
extern "C" void kernel_launch(void* const* d_in, const int* in_sizes, int n_in,
                              void* d_out, int out_size, void* d_ws, size_t ws_size,
                              hipStream_t stream) {
  (void)in_sizes; (void)n_in; (void)out_size; (void)ws_size;

  const float*     x        = (const float*)d_in[0];
  const long long* ei       = (const long long*)d_in[1];  // int64 [2,E]
  const float*     W_gat    = (const float*)d_in[3];
  const float*     att_src  = (const float*)d_in[4];
  const float*     att_dst  = (const float*)d_in[5];
  const float*     b_gat    = (const float*)d_in[6];
  const float*     W_sage_l = (const float*)d_in[7];
  const float*     b_sage_l = (const float*)d_in[8];
  const float*     W_sage_r = (const float*)d_in[9];
  const float*     W_lin1   = (const float*)d_in[10];
  const float*     b_lin1   = (const float*)d_in[11];
  const float*     W_lin2   = (const float*)d_in[12];
  const float*     b_lin2   = (const float*)d_in[13];
  float*           out      = (float*)d_out;

  // workspace layout (bytes); h buffer recycled as agg after message pass
  char* ws = (char*)d_ws;
  size_t off = 0;
  float*    h     = (float*)(ws + off);  off += (size_t)NND * HC * 4;        // 81.92 MB (later: agg)
  float*    out1  = (float*)(ws + off);  off += (size_t)NND * HC * 4;        // 81.92 MB (x1 after relu)
  float*    a_src = (float*)(ws + off);  off += (size_t)NND * HEADS * 4;
  float*    a_dst = (float*)(ws + off);  off += (size_t)NND * HEADS * 4;
  unsigned* emax  = (unsigned*)(ws + off); off += (size_t)NND * HEADS * 4;
  float*    denom = (float*)(ws + off);  off += (size_t)NND * HEADS * 4;
  float*    deg   = (float*)(ws + off);  off += (size_t)NND * 4;
  float*    eexp  = (float*)(ws + off);  off += (size_t)NE * HEADS * 4;
  float*    x2    = (float*)(ws + off);  off += (size_t)NND * HID * 4;
  float*    hmid  = (float*)(ws + off);  off += (size_t)NND * (HID / 2) * 4;
  float*    agg   = h;   // alias: h dead after k_message

  const int ELEMS = NND * HC;            // 20,480,000
  const int BLK   = 256;

  // 1) init accumulators (out1 seeded with b_gat; aggregate then relu)
  k_init1<<<ELEMS / BLK, BLK, 0, stream>>>(out1, b_gat, emax, denom, deg);

  // 2) h = x @ W_gat   [10000,128]x[128,2048]  (fp32 WMMA)
  k_wmma_gemm<INC, HC><<<dim3(NND / 16, (HC / 64) / 8), 256, 0, stream>>>(
      x, W_gat, nullptr, nullptr, nullptr, h, 0);

  // 3) attention coefficients
  k_attn<<<(NND + 7) / 8, 256, 0, stream>>>(h, att_src, att_dst, a_src, a_dst);

  // 4) edge logits + segment max + degree
  k_logits<<<(NE * HEADS + BLK - 1) / BLK, BLK, 0, stream>>>(ei, a_src, a_dst, eexp, emax, deg);

  // 5) exp + denominator
  k_exp<<<(NE * HEADS + BLK - 1) / BLK, BLK, 0, stream>>>(ei, eexp, emax, denom);

  // 6) message scatter: out1[dst] += alpha * h[src]
  k_message<<<(NE * HEADS) / 8, 256, 0, stream>>>(ei, h, eexp, denom, out1);

  // 7) x1 = relu(out1); zero agg
  k_relu_zero<<<ELEMS / BLK, BLK, 0, stream>>>(out1, agg);

  // 8) SAGE neighbor sum
  k_sage_scatter<<<NE / 8, 256, 0, stream>>>(ei, out1, agg);

  // 9) mean by degree
  k_sagediv<<<ELEMS / BLK, BLK, 0, stream>>>(agg, deg);

  // 10) x2 = relu(agg @ W_sage_l + x1 @ W_sage_r + b_sage_l)   (fused fp32 WMMA)
  k_wmma_gemm<HC, HID><<<dim3(NND / 16, 1), (HID / 64) * 32, 0, stream>>>(
      agg, W_sage_l, out1, W_sage_r, b_sage_l, x2, 1);

  // 11) hmid = relu(x2 @ W_lin1 + b_lin1)   [10000,256]x[256,128]
  k_wmma_gemm<HID, HID / 2><<<dim3(NND / 16, 1), ((HID / 2) / 64) * 32, 0, stream>>>(
      x2, W_lin1, nullptr, nullptr, b_lin1, hmid, 1);

  // 12) out = hmid @ W_lin2 + b_lin2
  k_mlp2<<<(NND + BLK - 1) / BLK, BLK, 0, stream>>>(hmid, W_lin2, b_lin2, out);
}